// SphericalHarmonicsLighting_51290499449216
// MI455X (gfx1250) — compile-verified
//
#include <hip/hip_runtime.h>

// Problem constants (fixed by the reference): B=16, V=2^18
#define SH_B 16
#define SH_V 262144u

typedef __attribute__((ext_vector_type(2))) float v2f;
typedef __attribute__((ext_vector_type(8))) float v8f;

// T (16x9) padded to 16x12 (K padded to multiple of 4 for 3x WMMA K=4 chunks).
// c = {0.429043, 0.511664, 0.743125, 0.886227, 0.247708}
__device__ const float Tpad[16][12] = {
    {0.f,0.f,0.f,0.f,  0.f,0.f,0.f,0.f,  0.429043f,0.f,0.f,0.f},   // r0:  col8= c0
    {0.f,0.f,0.f,0.f,  0.429043f,0.f,0.f,0.f,  0.f,0.f,0.f,0.f},   // r1:  col4= c0
    {0.f,0.f,0.f,0.f,  0.f,0.f,0.f,0.429043f,  0.f,0.f,0.f,0.f},   // r2:  col7= c0
    {0.f,0.f,0.f,0.511664f,  0.f,0.f,0.f,0.f,  0.f,0.f,0.f,0.f},   // r3:  col3= c1
    {0.f,0.f,0.f,0.f,  0.429043f,0.f,0.f,0.f,  0.f,0.f,0.f,0.f},   // r4:  col4= c0
    {0.f,0.f,0.f,0.f,  0.f,0.f,0.f,0.f,  -0.429043f,0.f,0.f,0.f},  // r5:  col8=-c0
    {0.f,0.f,0.f,0.f,  0.f,0.429043f,0.f,0.f,  0.f,0.f,0.f,0.f},   // r6:  col5= c0
    {0.f,0.511664f,0.f,0.f,  0.f,0.f,0.f,0.f,  0.f,0.f,0.f,0.f},   // r7:  col1= c1
    {0.f,0.f,0.f,0.f,  0.f,0.f,0.f,0.429043f,  0.f,0.f,0.f,0.f},   // r8:  col7= c0
    {0.f,0.f,0.f,0.f,  0.f,0.429043f,0.f,0.f,  0.f,0.f,0.f,0.f},   // r9:  col5= c0
    {0.f,0.f,0.f,0.f,  0.f,0.f,0.743125f,0.f,  0.f,0.f,0.f,0.f},   // r10: col6= c2
    {0.f,0.f,0.511664f,0.f,  0.f,0.f,0.f,0.f,  0.f,0.f,0.f,0.f},   // r11: col2= c1
    {0.f,0.f,0.f,0.511664f,  0.f,0.f,0.f,0.f,  0.f,0.f,0.f,0.f},   // r12: col3= c1
    {0.f,0.511664f,0.f,0.f,  0.f,0.f,0.f,0.f,  0.f,0.f,0.f,0.f},   // r13: col1= c1
    {0.f,0.f,0.511664f,0.f,  0.f,0.f,0.f,0.f,  0.f,0.f,0.f,0.f},   // r14: col2= c1
    {0.886227f,0.f,0.f,0.f,  0.f,0.f,-0.247708f,0.f,  0.f,0.f,0.f,0.f}, // r15: col0=c3, col6=-c4
};

__global__ __launch_bounds__(256)
void sh_lighting_kernel(const float* __restrict__ normals,
                        const float* __restrict__ sh,   // 27 = 3 channels x 9 coeffs
                        float* __restrict__ out) {
    // Ml[i*12 + c*4 + k] = M[c][i][k]  (48 floats)
    __shared__ float Ml[48];

    // ---- Index algebra (block-uniform parts computed from blockIdx -> SALU)
    const unsigned tid = blockIdx.x * 256u + threadIdx.x;           // (b,v) flat
    const unsigned b   = blockIdx.x >> 10;                          // tid >> 18
    const unsigned v   = ((blockIdx.x & 1023u) << 8) + threadIdx.x; // tid & (V-1)
    const unsigned i   = (blockIdx.x >> 8) & 3u;                    // v >> 16, block-uniform
    const unsigned u   = (4u * v) & (SH_V - 1u);                    // (4v) mod V, multiple of 4

    const float* __restrict__ Nb = normals + (size_t)b * (3u * SH_V);

    // ---- Issue hot-path global loads FIRST: their latency overlaps the WMMA
    //      preamble below (waits land at first use, after the barrier).
    const float nx = Nb[3u * v + 0u];
    const float ny = Nb[3u * v + 1u];
    const float nz = Nb[3u * v + 2u];
    // gather normals u..u+3: 12 contiguous floats, 16B-aligned (u % 4 == 0)
    const float4* __restrict__ P = reinterpret_cast<const float4*>(Nb + 3u * u);
    const float4 q0 = P[0];  // n(u).xyz,  n(u+1).x
    const float4 q1 = P[1];  // n(u+1).yz, n(u+2).xy
    const float4 q2 = P[2];  // n(u+2).z,  n(u+3).xyz

    // ---- Stage 1: wave 0 computes M = (T @ sh.reshape(3,9).T) via WMMA f32 16x16x4
    if (threadIdx.x < 32u) {
        const int lane = (int)threadIdx.x;
        const int r    = lane & 15;     // A-matrix row (M dim)
        const int half = lane >> 4;     // 0: lanes 0-15, 1: lanes 16-31
        const int n    = lane & 15;     // B/D column (N dim)
        const int ncl  = (n < 3) ? n : 2;  // clamped channel for always-in-bounds loads

        v8f acc = {};
        #pragma unroll
        for (int chunk = 0; chunk < 3; ++chunk) {
            const int k0 = chunk * 4;
            // A 16x4 layout: lanes0-15 hold K=k0+0/+1 in VGPR0/1; lanes16-31 K=k0+2/+3
            const int ka = k0 + (half ? 2 : 0);
            v2f a;
            a.x = Tpad[r][ka];
            a.y = Tpad[r][ka + 1];
            // B 4x16 layout (row-striped across lanes): VGPR0 -> K rows {0,2}, VGPR1 -> {1,3}
            const int kb0 = k0 + (half ? 2 : 0);
            const int kb1 = kb0 + 1;
            // Branchless: clamp indices so loads are unconditional + in-bounds,
            // zero invalid entries with a value select (v_cndmask, no exec branch).
            const float sx = sh[ncl * 9 + ((kb0 < 9) ? kb0 : 8)];
            const float sy = sh[ncl * 9 + ((kb1 < 9) ? kb1 : 8)];
            v2f bv;
            bv.x = (n < 3 && kb0 < 9) ? sx : 0.0f;
            bv.y = (n < 3 && kb1 < 9) ? sy : 0.0f;
            // D(16x16) += A(16x4) x B(4x16);  D[r][n] = M_flat[n][r]
            acc = __builtin_amdgcn_wmma_f32_16x16x4_f32(
                false, a, false, bv, (short)0, acc, false, false);
        }
        if (n < 3) {  // only columns 0..2 carry RGB channels
            #pragma unroll
            for (int p = 0; p < 8; ++p) {
                const int rr = p + 8 * half;   // D row = flattened M index i*4+k
                const int ii = rr >> 2;
                const int kk = rr & 3;
                Ml[ii * 12 + n * 4 + kk] = acc[p];
            }
        }
    }
    __syncthreads();

    // ---- Stage 2: bandwidth-bound per-vertex evaluation
    // Block-uniform rows of M for this i (LDS broadcast)
    float m[12];
    #pragma unroll
    for (int t = 0; t < 12; ++t) m[t] = Ml[i * 12u + t];

    // G[k] = sum_j N4[v][j] * N4[u+j][k]
    const float G0 = nx * q0.x + ny * q0.w + nz * q1.z + q2.y;
    const float G1 = nx * q0.y + ny * q1.x + nz * q1.w + q2.z;
    const float G2 = nx * q0.z + ny * q1.y + nz * q2.x + q2.w;
    const float G3 = nx + ny + nz + 1.0f;

    // E[b,v,c] = M[c][i][:] . G
    const float o0 = m[0] * G0 + m[1] * G1 + m[2]  * G2 + m[3]  * G3;
    const float o1 = m[4] * G0 + m[5] * G1 + m[6]  * G2 + m[7]  * G3;
    const float o2 = m[8] * G0 + m[9] * G1 + m[10] * G2 + m[11] * G3;

    // streaming output (never re-read): nontemporal to keep normals L2-resident
    float* __restrict__ O = out + (size_t)tid * 3u;
    __builtin_nontemporal_store(o0, O + 0);
    __builtin_nontemporal_store(o1, O + 1);
    __builtin_nontemporal_store(o2, O + 2);
}

extern "C" void kernel_launch(void* const* d_in, const int* in_sizes, int n_in,
                              void* d_out, int out_size, void* d_ws, size_t ws_size,
                              hipStream_t stream) {
    (void)in_sizes; (void)n_in; (void)out_size; (void)d_ws; (void)ws_size;
    // d_in[0] = light (3)   -- unused by the reference computation
    // d_in[1] = normals (B*V*3) float32
    // d_in[2] = sh_coeffs (27) float32
    const float* normals = (const float*)d_in[1];
    const float* sh      = (const float*)d_in[2];
    float*       out     = (float*)d_out;

    const unsigned blocks = (SH_B * SH_V) / 256u;  // 16384
    sh_lighting_kernel<<<blocks, 256, 0, stream>>>(normals, sh, out);
}